// IntegerQuantization_58866821759056
// MI455X (gfx1250) — compile-verified
//
#include <hip/hip_runtime.h>

// Problem constants (from reference setup_inputs)
#define B_DIM     32
#define C_DIM     256
#define HW_DIM    4096                 // 64*64
#define NUM_BINS  256
#define N_TOTAL   (B_DIM * C_DIM * HW_DIM)   // 33,554,432
#define N_PER_CH  (B_DIM * HW_DIM)           // 131,072 (power of two -> exact /n)
#define MOM       0.99f

// gfx1250 async global->LDS path (guarded: falls back to direct loads)
#if defined(__HIP_DEVICE_COMPILE__) && \
    __has_builtin(__builtin_amdgcn_global_load_async_to_lds_b128) && \
    __has_builtin(__builtin_amdgcn_s_wait_asynccnt)
#define HAVE_ASYNC_LDS 1
#else
#define HAVE_ASYNC_LDS 0
#endif

#if HAVE_ASYNC_LDS
typedef int v4i_t __attribute__((ext_vector_type(4)));
typedef __attribute__((address_space(1))) v4i_t* GlobalV4;
typedef __attribute__((address_space(3))) v4i_t* LdsV4;

__device__ __forceinline__ void issue_async_b128(const float4* gsrc, float4* lds_dst) {
    __builtin_amdgcn_global_load_async_to_lds_b128(
        (GlobalV4)(float4*)gsrc, (LdsV4)lds_dst, /*offset=*/0, /*cpol=*/0);
}

template <int PENDING>
__device__ __forceinline__ void wait_async() {
    __builtin_amdgcn_s_wait_asynccnt(PENDING);
}
#endif

// ---------------------------------------------------------------------------
// Kernel 0: zero the global histogram workspace (C*NUM_BINS u32 = 256 KB)
// ---------------------------------------------------------------------------
__global__ void zero_hist_kernel(unsigned int* __restrict__ hist) {
    hist[blockIdx.x * blockDim.x + threadIdx.x] = 0u;
}

// ---------------------------------------------------------------------------
// Kernel 1: quantize + per-channel histogram.
// One block per (b,c) slice of 4096 contiguous floats => whole block hits a
// single channel => one 256-bin LDS histogram (ds_add_u32), merged once.
// Data is streamed via async global->LDS (double-buffered, per-lane slots,
// per-wave s_wait_asynccnt) when available.
// ---------------------------------------------------------------------------
__global__ __launch_bounds__(256) void quant_hist_kernel(
    const float* __restrict__ x,
    float* __restrict__ xq,
    unsigned int* __restrict__ hist)
{
    __shared__ unsigned int shist[NUM_BINS];
#if HAVE_ASYNC_LDS
    __shared__ float4 stage[2][256];   // 8 KB staging, each lane owns one slot
#endif
    const int tid = threadIdx.x;
    shist[tid] = 0u;
    __syncthreads();

    const long long base = (long long)blockIdx.x * HW_DIM;
    const float4* __restrict__ xv = (const float4*)(x + base);
    float4* __restrict__ qv = (float4*)(xq + base);

    // process one float4 chunk (256 lanes x 16B = 4KB per chunk, 4 chunks)
    auto process = [&](int k, float4 v) {
        float4 r;
        r.x = rintf(fminf(v.x, 255.0f));   // round-half-even == jnp.round
        r.y = rintf(fminf(v.y, 255.0f));
        r.z = rintf(fminf(v.z, 255.0f));
        r.w = rintf(fminf(v.w, 255.0f));
        qv[k * 256 + tid] = r;
        const int b0 = min(max((int)r.x, 0), NUM_BINS - 1);
        const int b1 = min(max((int)r.y, 0), NUM_BINS - 1);
        const int b2 = min(max((int)r.z, 0), NUM_BINS - 1);
        const int b3 = min(max((int)r.w, 0), NUM_BINS - 1);
        atomicAdd(&shist[b0], 1u);
        atomicAdd(&shist[b1], 1u);
        atomicAdd(&shist[b2], 1u);
        atomicAdd(&shist[b3], 1u);
    };

#if HAVE_ASYNC_LDS
    // double-buffered async pipeline: chunk k+1 streams in while k is consumed
    issue_async_b128(&xv[0 * 256 + tid], &stage[0][tid]);
    issue_async_b128(&xv[1 * 256 + tid], &stage[1][tid]);

    wait_async<1>();                 // oldest (chunk 0) landed
    process(0, stage[0][tid]);       // ds_load_b128 from own slot
    issue_async_b128(&xv[2 * 256 + tid], &stage[0][tid]);

    wait_async<1>();
    process(1, stage[1][tid]);
    issue_async_b128(&xv[3 * 256 + tid], &stage[1][tid]);

    wait_async<1>();
    process(2, stage[0][tid]);

    wait_async<0>();
    process(3, stage[1][tid]);
#else
#pragma unroll
    for (int k = 0; k < 4; ++k) {
        __builtin_prefetch(&xv[k * 256 + tid + 256], 0, 0);  // global_prefetch_b8
        process(k, xv[k * 256 + tid]);
    }
#endif
    __syncthreads();

    const int c = blockIdx.x & (C_DIM - 1);   // slice index = b*C + c
    const unsigned int cnt = shist[tid];
    if (cnt != 0u) {
        atomicAdd(&hist[c * NUM_BINS + tid], cnt);
    }
}

// ---------------------------------------------------------------------------
// Kernel 2: build p_new row for this block's channel in LDS, then gather.
// p_new[c,bin] = p*0.99 + (hist/131072)*(1-0.99); /131072 is exact (pow2).
// Gather becomes an LDS read; xq is re-read from d_out (written by kernel 1).
// ---------------------------------------------------------------------------
__global__ __launch_bounds__(256) void pnew_gather_kernel(
    const float* __restrict__ xq,
    const float* __restrict__ estimated_p,
    const unsigned int* __restrict__ hist,
    float* __restrict__ px)
{
    __shared__ float prow[NUM_BINS];
    const int tid = threadIdx.x;
    const int c = blockIdx.x & (C_DIM - 1);

    const float one_minus = 1.0f - MOM;               // matches reference fp32 math
    const unsigned int h = hist[c * NUM_BINS + tid];
    prow[tid] = estimated_p[c * NUM_BINS + tid] * MOM
              + ((float)h * (1.0f / (float)N_PER_CH)) * one_minus;
    __syncthreads();

    const long long base = (long long)blockIdx.x * HW_DIM;
    const float4* __restrict__ qv = (const float4*)(xq + base);
    float4* __restrict__ pv = (float4*)(px + base);

#pragma unroll
    for (int k = 0; k < 4; ++k) {
        const int i = k * 256 + tid;
        __builtin_prefetch(&qv[i + 256], 0, 0);
        float4 v = qv[i];
        float4 p;
        p.x = prow[min(max((int)v.x, 0), NUM_BINS - 1)];
        p.y = prow[min(max((int)v.y, 0), NUM_BINS - 1)];
        p.z = prow[min(max((int)v.z, 0), NUM_BINS - 1)];
        p.w = prow[min(max((int)v.w, 0), NUM_BINS - 1)];
        pv[i] = p;
    }
}

// ---------------------------------------------------------------------------
extern "C" void kernel_launch(void* const* d_in, const int* in_sizes, int n_in,
                              void* d_out, int out_size, void* d_ws, size_t ws_size,
                              hipStream_t stream) {
    const float* x  = (const float*)d_in[0];          // (32,256,64,64) f32
    const float* ep = (const float*)d_in[1];          // (256,256) f32
    float* xq = (float*)d_out;                        // output 0: x_q
    float* px = (float*)d_out + N_TOTAL;              // output 1: p_x
    unsigned int* hist = (unsigned int*)d_ws;         // 256 KB scratch

    const int n_slices = B_DIM * C_DIM;               // 8192 blocks

    zero_hist_kernel<<<C_DIM, NUM_BINS, 0, stream>>>(hist);
    quant_hist_kernel<<<n_slices, 256, 0, stream>>>(x, xq, hist);
    pnew_gather_kernel<<<n_slices, 256, 0, stream>>>(xq, ep, hist, px);
}